// DPAtt2Attention_11304353923737
// MI455X (gfx1250) — compile-verified
//
#include <hip/hip_runtime.h>
#include <hip/hip_bf16.h>

typedef __attribute__((ext_vector_type(2))) float v2f;
typedef __attribute__((ext_vector_type(8))) float v8f;

#define LSEQ   200
#define LP     208           // padded L (13 * 16)
#define LL     (LP * LP)     // 43264
#define NT     13            // LP / 16
#define DMODEL 64
#define BATCH  256
#define NROWS  (BATCH * LSEQ)     // 51200
#define BSTRIDE (LP * DMODEL)     // 13312 floats: per-batch stride of S2p and Vt
#define INV_SQRT_HEAD 0.17677669529663687f  // 1/sqrt(32)

static __device__ __forceinline__ v8f wmma_f32(v2f a, v2f b, v8f c) {
  // V_WMMA_F32_16X16X4_F32 : D(16x16 f32) = A(16x4 f32) * B(4x16 f32) + C
  return __builtin_amdgcn_wmma_f32_16x16x4_f32(false, a, false, b, (short)0, c,
                                               false, false);
}

// ---------------------------------------------------------------------------
// K0: zero the pad rows (200..207) of the batch-padded S2 buffer.
// ---------------------------------------------------------------------------
__global__ void pad_zero_kernel(float* __restrict__ S2p) {
  const int b = blockIdx.x;
  float* p = S2p + (size_t)b * BSTRIDE + LSEQ * DMODEL;
  for (int i = threadIdx.x; i < (LP - LSEQ) * DMODEL; i += 256) p[i] = 0.f;
}

// ---------------------------------------------------------------------------
// K1: S2p[b][r][k] = (x@Wq^T+bq)^2  (batch-padded rows) and
//     Vt[b][n][r]  =  x@Wv^T+bv     (transposed, row stride LP).
// grid = (NROWS/16, 8): y<4 -> Q col-tile y; y>=4 -> V col-tile y-4. 1 wave.
// ---------------------------------------------------------------------------
__global__ void proj_sq_kernel(const float* __restrict__ x,
                               const float* __restrict__ Wq, const float* __restrict__ bq,
                               const float* __restrict__ Wv, const float* __restrict__ bv,
                               float* __restrict__ S2p, float* __restrict__ Vt) {
  const int lane  = threadIdx.x;
  const int mt    = blockIdx.x;
  const int yy    = blockIdx.y;
  const bool isQ  = (yy < 4);
  const int ct    = isQ ? yy : (yy - 4);
  const float* W    = isQ ? Wq : Wv;
  const float* bias = isQ ? bq : bv;

  const int ml = lane & 15;
  const int hi = lane >> 4;
  const int kh = hi * 2;
  const float* pa = x + (size_t)(mt * 16 + ml) * DMODEL;
  const float* pw = W + (size_t)(ct * 16 + ml) * DMODEL;

  v8f acc = {0.f, 0.f, 0.f, 0.f, 0.f, 0.f, 0.f, 0.f};
#pragma unroll
  for (int kb = 0; kb < DMODEL; kb += 4) {
    v2f a  = *(const v2f*)(pa + kb + kh);
    v2f bb = *(const v2f*)(pw + kb + kh);
    acc = wmma_f32(a, bb, acc);
  }
#pragma unroll
  for (int i = 0; i < 8; ++i) {
    const int m  = mt * 16 + i + hi * 8;       // global row
    const int gb = m / LSEQ;
    const int r  = m - gb * LSEQ;
    const int n  = ct * 16 + ml;
    const float v = acc[i] + bias[n];
    if (isQ)
      S2p[(size_t)gb * BSTRIDE + r * DMODEL + n] = v * v;
    else
      Vt[(size_t)gb * BSTRIDE + n * LP + r] = v;
  }
}

// ---------------------------------------------------------------------------
// K2: L_kernel[b] = S2_b @ S2_b^T.  One wave computes a 16x208 row of tiles:
// A fragment loaded once per K-step, reused for 13 WMMAs. grid = (BATCH, 13).
// ---------------------------------------------------------------------------
__global__ void lkernel_kernel(const float* __restrict__ S2p, float* __restrict__ Lk) {
  const int lane = threadIdx.x;
  const int b = blockIdx.x, mt = blockIdx.y;
  const int ml = lane & 15;
  const int hi = lane >> 4;
  const int kh = hi * 2;
  const float* Sb  = S2p + (size_t)b * BSTRIDE;
  const float* pa  = Sb + (size_t)(mt * 16 + ml) * DMODEL;
  const float* pb0 = Sb + (size_t)ml * DMODEL;

  v8f acc[NT];
#pragma unroll
  for (int t = 0; t < NT; ++t) acc[t] = (v8f){0.f,0.f,0.f,0.f,0.f,0.f,0.f,0.f};

#pragma unroll
  for (int kb = 0; kb < DMODEL; kb += 4) {
    const v2f a = *(const v2f*)(pa + kb + kh);
#pragma unroll
    for (int t = 0; t < NT; ++t) {
      const v2f bb = *(const v2f*)(pb0 + t * 16 * DMODEL + kb + kh);
      acc[t] = wmma_f32(a, bb, acc[t]);
    }
  }
  float* pst = Lk + (size_t)b * LL + (size_t)(mt * 16 + hi * 8) * LP + ml;
#pragma unroll
  for (int i = 0; i < 8; ++i)
#pragma unroll
    for (int t = 0; t < NT; ++t) pst[i * LP + t * 16] = acc[t][i];
}

// ---------------------------------------------------------------------------
// K3: A[b] = C_kernel[ids x ids] + 1e-4*I  (padded region = identity).
// ---------------------------------------------------------------------------
__global__ void gather_kernel(const int* __restrict__ ids,
                              const float* __restrict__ C,
                              float* __restrict__ A) {
  const int l = blockIdx.x;     // row 0..207
  const int b = blockIdx.y;
  const int m = threadIdx.x;    // col
  if (m >= LP) return;
  const int* idb = ids + b * LSEQ;
  float v;
  if (l < LSEQ && m < LSEQ) {
    const long r = idb[l];
    const long c = idb[m];
    v = C[r * 10000 + c] + ((l == m) ? 1e-4f : 0.f);
  } else {
    v = (l == m) ? 1.f : 0.f;
  }
  A[(size_t)b * LL + l * LP + m] = v;
}

// ---------------------------------------------------------------------------
// K4: in-place Gauss-Jordan inverse of the 200x200 block, fully in LDS.
// 200*201*4 = 160,800 B dynamic LDS (CDNA5: 320 KB per workgroup).
// Stride 201 -> bank stride 9, coprime with 64 banks: conflict-free columns.
// SPD input (0.1*FF^T + (1+1e-4)I) => pivoting-free GJ is numerically safe.
// ---------------------------------------------------------------------------
__global__ void gj_inverse_kernel(float* __restrict__ A) {
  extern __shared__ float As[];
  __shared__ float ipiv_s;
  const int S = 201;
  const int tid = threadIdx.x;
  float* Ag = A + (size_t)blockIdx.x * LL;

  for (int idx = tid; idx < LSEQ * LSEQ; idx += 256) {
    const int r = idx / LSEQ, c = idx % LSEQ;
    As[r * S + c] = Ag[r * LP + c];
  }
  __syncthreads();

  for (int p = 0; p < LSEQ; ++p) {
    if (tid == 0) ipiv_s = 1.0f / As[p * S + p];
    __syncthreads();
    const float ipiv = ipiv_s;
    for (int j = tid; j < LSEQ; j += 256)
      if (j != p) As[p * S + j] *= ipiv;
    if (tid == 0) As[p * S + p] = ipiv;
    __syncthreads();
    if (tid < LSEQ && tid != p) {
      const int i = tid;
      const float f = As[i * S + p];
      const float nf = -f;
      for (int j = 0; j < LSEQ; ++j)
        As[i * S + j] = fmaf(nf, As[p * S + j], As[i * S + j]);
      As[i * S + p] = nf * ipiv;
    }
    __syncthreads();
  }

  for (int idx = tid; idx < LSEQ * LSEQ; idx += 256) {
    const int r = idx / LSEQ, c = idx % LSEQ;
    Ag[r * LP + c] = As[r * S + c];
  }
}

// ---------------------------------------------------------------------------
// K5: T[b] = L_kernel[b] @ C_inv[b].  C_inv is symmetric, so the B operand is
// read row-wise (contiguous b64 loads).  16x208 row of tiles per wave.
// grid = (BATCH, 13).
// ---------------------------------------------------------------------------
__global__ void gemmT_kernel(const float* __restrict__ Lk,
                             const float* __restrict__ Ainv,
                             float* __restrict__ T) {
  const int lane = threadIdx.x;
  const int b = blockIdx.x, mt = blockIdx.y;
  const int ml = lane & 15;
  const int hi = lane >> 4;
  const int kh = hi * 2;
  const float* pa  = Lk   + (size_t)b * LL + (size_t)(mt * 16 + ml) * LP;
  const float* pb0 = Ainv + (size_t)b * LL + (size_t)ml * LP;   // row n (symmetry)

  v8f acc[NT];
#pragma unroll
  for (int t = 0; t < NT; ++t) acc[t] = (v8f){0.f,0.f,0.f,0.f,0.f,0.f,0.f,0.f};

#pragma unroll 2
  for (int kb = 0; kb < LSEQ; kb += 4) {
    const v2f a = *(const v2f*)(pa + kb + kh);
#pragma unroll
    for (int t = 0; t < NT; ++t) {
      const v2f bb = *(const v2f*)(pb0 + t * 16 * LP + kb + kh);
      acc[t] = wmma_f32(a, bb, acc[t]);
    }
  }
  float* pst = T + (size_t)b * LL + (size_t)(mt * 16 + hi * 8) * LP + ml;
#pragma unroll
  for (int i = 0; i < 8; ++i)
#pragma unroll
    for (int t = 0; t < NT; ++t) pst[i * LP + t * 16] = acc[t][i];
}

// ---------------------------------------------------------------------------
// K6: sub-determinant scores + softmax -> attn (written into Lk buffer).
// grid = BATCH, block = 256 (8 waves; one row per wave, wave32 shfl reduce).
// ---------------------------------------------------------------------------
__global__ void score_softmax_kernel(const float* __restrict__ T,
                                     const float* __restrict__ mask,
                                     float* __restrict__ attn) {
  __shared__ float diag[LSEQ];
  __shared__ float red[8];
  __shared__ float denom_s;
  const int tid  = threadIdx.x;
  const int lane = tid & 31;
  const int wid  = tid >> 5;
  const int b = blockIdx.x;
  const float* Tb = T + (size_t)b * LL;
  const float* Mb = mask + (size_t)b * LSEQ * LSEQ;
  float* Ab = attn + (size_t)b * LL;

  for (int l = tid; l < LSEQ; l += 256) diag[l] = Tb[l * LP + l];
  __syncthreads();

  float part = 0.f;
  for (int idx = tid; idx < LSEQ * LSEQ; idx += 256) {
    const int l = idx / LSEQ, m = idx % LSEQ;
    if (m > l) part += diag[l] * diag[m] - Tb[l * LP + m] * Tb[m * LP + l];
  }
  for (int off = 16; off >= 1; off >>= 1) part += __shfl_xor(part, off, 32);
  if (lane == 0) red[wid] = part;
  __syncthreads();
  if (tid == 0) {
    float d = 0.f;
    for (int i = 0; i < 8; ++i) d += red[i];
    denom_s = fmaxf(d, 1e-9f);
  }
  __syncthreads();
  const float inv_denom = 1.0f / denom_s;

  for (int r = wid; r < LSEQ; r += 8) {
    float sv[7];
    float mx = -3.0e38f;
#pragma unroll
    for (int t = 0; t < 7; ++t) {
      const int m = lane + t * 32;
      float s = -3.0e38f;
      if (m < LSEQ) {
        const float sub = diag[r] * diag[m] - Tb[r * LP + m] * Tb[m * LP + r];
        s = -(sub * inv_denom + ((m == r) ? diag[r] : 0.f)) * INV_SQRT_HEAD
            + Mb[r * LSEQ + m];
      }
      sv[t] = s;
      mx = fmaxf(mx, s);
    }
    for (int off = 16; off >= 1; off >>= 1) mx = fmaxf(mx, __shfl_xor(mx, off, 32));
    float sum = 0.f;
#pragma unroll
    for (int t = 0; t < 7; ++t) {
      const int m = lane + t * 32;
      if (m < LSEQ) { sv[t] = __expf(sv[t] - mx); sum += sv[t]; }
      else sv[t] = 0.f;
    }
    for (int off = 16; off >= 1; off >>= 1) sum += __shfl_xor(sum, off, 32);
    const float inv = 1.0f / sum;
#pragma unroll
    for (int t = 0; t < 7; ++t) {
      const int m = lane + t * 32;
      if (m < LP) Ab[r * LP + m] = (m < LSEQ) ? sv[t] * inv : 0.f;
    }
  }
}

// ---------------------------------------------------------------------------
// K7: ctx[b] = attn[b] @ V[b].  B operand from transposed Vt (contiguous).
// One wave computes 16x64 (4 accumulators).  grid = (BATCH, 13).
// ---------------------------------------------------------------------------
__global__ void gemm_ctx_kernel(const float* __restrict__ attn,
                                const float* __restrict__ Vt,
                                float* __restrict__ ctx) {
  const int lane = threadIdx.x;
  const int b = blockIdx.x, mt = blockIdx.y;
  const int ml = lane & 15;
  const int hi = lane >> 4;
  const int kh = hi * 2;
  const float* pa  = attn + (size_t)b * LL + (size_t)(mt * 16 + ml) * LP;
  const float* pb0 = Vt   + (size_t)b * BSTRIDE + (size_t)ml * LP;

  v8f acc[4];
#pragma unroll
  for (int t = 0; t < 4; ++t) acc[t] = (v8f){0.f,0.f,0.f,0.f,0.f,0.f,0.f,0.f};

#pragma unroll 2
  for (int kb = 0; kb < LSEQ; kb += 4) {
    const v2f a = *(const v2f*)(pa + kb + kh);
#pragma unroll
    for (int t = 0; t < 4; ++t) {
      const v2f bb = *(const v2f*)(pb0 + t * 16 * LP + kb + kh);
      acc[t] = wmma_f32(a, bb, acc[t]);
    }
  }
  float* Cb = ctx + (size_t)b * LSEQ * DMODEL;
#pragma unroll
  for (int i = 0; i < 8; ++i) {
    const int m = mt * 16 + i + hi * 8;
    if (m < LSEQ) {
#pragma unroll
      for (int t = 0; t < 4; ++t) Cb[m * DMODEL + t * 16 + ml] = acc[t][i];
    }
  }
}

// ---------------------------------------------------------------------------
// K8: out = LayerNorm(ctx @ Wd^T + bd + x).  grid = NROWS/16, block = 128.
// ---------------------------------------------------------------------------
__global__ void out_ln_kernel(const float* __restrict__ ctx,
                              const float* __restrict__ x,
                              const float* __restrict__ Wd, const float* __restrict__ bd,
                              const float* __restrict__ gamma, const float* __restrict__ beta,
                              float* __restrict__ out) {
  __shared__ float ybuf[16][DMODEL + 1];
  __shared__ float mu_s[16], rstd_s[16];
  const int tid  = threadIdx.x;
  const int lane = tid & 31;
  const int w    = tid >> 5;
  const int r0   = blockIdx.x * 16;
  const int ml = lane & 15;
  const int hi = lane >> 4;
  const int kh = hi * 2;
  const float* pa = ctx + (size_t)(r0 + ml) * DMODEL;
  const float* pw = Wd  + (size_t)(w * 16 + ml) * DMODEL;

  v8f acc = {0.f, 0.f, 0.f, 0.f, 0.f, 0.f, 0.f, 0.f};
#pragma unroll
  for (int kb = 0; kb < DMODEL; kb += 4) {
    v2f a  = *(const v2f*)(pa + kb + kh);
    v2f bb = *(const v2f*)(pw + kb + kh);
    acc = wmma_f32(a, bb, acc);
  }
#pragma unroll
  for (int i = 0; i < 8; ++i) {
    const int m = i + hi * 8;
    const int n = w * 16 + ml;
    ybuf[m][n] = acc[i] + bd[n] + x[(r0 + m) * DMODEL + n];
  }
  __syncthreads();
  if (tid < 16) {
    float mu = 0.f;
    for (int j = 0; j < DMODEL; ++j) mu += ybuf[tid][j];
    mu *= (1.0f / DMODEL);
    float var = 0.f;
    for (int j = 0; j < DMODEL; ++j) {
      const float d = ybuf[tid][j] - mu;
      var += d * d;
    }
    var *= (1.0f / DMODEL);
    mu_s[tid] = mu;
    rstd_s[tid] = rsqrtf(var + 1e-12f);
  }
  __syncthreads();
  for (int idx = tid; idx < 16 * DMODEL; idx += 128) {
    const int m = idx >> 6, n = idx & 63;
    out[(r0 + m) * DMODEL + n] =
        gamma[n] * (ybuf[m][n] - mu_s[m]) * rstd_s[m] + beta[n];
  }
}

// ---------------------------------------------------------------------------
extern "C" void kernel_launch(void* const* d_in, const int* in_sizes, int n_in,
                              void* d_out, int out_size, void* d_ws, size_t ws_size,
                              hipStream_t stream) {
  (void)in_sizes; (void)n_in; (void)out_size; (void)ws_size;
  const float* x     = (const float*)d_in[0];
  const float* mask  = (const float*)d_in[1];
  const int*   ids   = (const int*)d_in[2];
  const float* Ck    = (const float*)d_in[3];
  const float* Wq    = (const float*)d_in[4];
  const float* bq    = (const float*)d_in[5];
  const float* Wv    = (const float*)d_in[6];
  const float* bv    = (const float*)d_in[7];
  const float* Wd    = (const float*)d_in[8];
  const float* bd    = (const float*)d_in[9];
  const float* gamma = (const float*)d_in[10];
  const float* beta  = (const float*)d_in[11];
  float* out = (float*)d_out;

  float* ws  = (float*)d_ws;
  float* S2p = ws;                                   // 256*208*64 (pad rows zeroed)
  float* Vt  = S2p + (size_t)BATCH * BSTRIDE;        // 256*64*208 (V transposed)
  float* Wlk = Vt  + (size_t)BATCH * BSTRIDE;        // 256*208*208 (L_kernel -> attn)
  float* WA  = Wlk + (size_t)BATCH * LL;             // 256*208*208 (C_sub -> C_inv)
  float* WT  = WA  + (size_t)BATCH * LL;             // 256*208*208 (T)
  float* CTX = S2p;                                  // reuse S2p region (fits: 200*64 < 208*64)

  pad_zero_kernel<<<dim3(BATCH), 256, 0, stream>>>(S2p);
  proj_sq_kernel<<<dim3(NROWS / 16, 8), 32, 0, stream>>>(x, Wq, bq, Wv, bv, S2p, Vt);
  lkernel_kernel<<<dim3(BATCH, NT), 32, 0, stream>>>(S2p, Wlk);
  gather_kernel<<<dim3(LP, BATCH), 256, 0, stream>>>(ids, Ck, WA);
  gj_inverse_kernel<<<dim3(BATCH), 256, 200 * 201 * 4, stream>>>(WA);
  gemmT_kernel<<<dim3(BATCH, NT), 32, 0, stream>>>(Wlk, WA, WT);
  score_softmax_kernel<<<dim3(BATCH), 256, 0, stream>>>(WT, mask, Wlk);
  gemm_ctx_kernel<<<dim3(BATCH, NT), 32, 0, stream>>>(Wlk, Vt, CTX);
  out_ln_kernel<<<dim3(NROWS / 16), 128, 0, stream>>>(CTX, x, Wd, bd, gamma, beta, out);
}